// model_arrival_68985764708960
// MI455X (gfx1250) — compile-verified
//
#include <hip/hip_runtime.h>

// EV-charging simulation, collapsed to cohorts.
// Pass 1: zero 128 bins in d_ws.
// Pass 2: histogram of arrival times (the only O(N) work; memory-bound:
//         32MB / 23.3 TB/s ~ 1.4us). float4 (global_load_b128) streaming,
//         per-wave LDS sub-histograms (ds_add), global atomic merge.
// Pass 3: single-wave cohort simulation, 95 steps x 128 bins; wave-wide
//         sums done with v_wmma_f32_16x16x4_f32 (A = data, B = ones).

#define T_STEPS 96
#define NBINS 128
#define WAVES_PER_BLOCK 8
#define HIST_BLOCK 256
#define HIST_GRID 2048

typedef float v2f __attribute__((ext_vector_type(2)));
typedef float v8f __attribute__((ext_vector_type(8)));

__global__ void zero_bins_kernel(int* __restrict__ bins) {
  int i = threadIdx.x;
  if (i < NBINS) bins[i] = 0;
}

__device__ __forceinline__ int clamp_bin(float x) {
  int b = (int)x;
  b = b < 0 ? 0 : b;
  b = b > (NBINS - 1) ? (NBINS - 1) : b;
  return b;
}

__global__ __launch_bounds__(HIST_BLOCK)
void hist_kernel(const float* __restrict__ arr, int n, int* __restrict__ bins) {
  __shared__ int h[WAVES_PER_BLOCK][NBINS];
  const int wave = threadIdx.x >> 5;
  for (int i = threadIdx.x; i < WAVES_PER_BLOCK * NBINS; i += blockDim.x)
    (&h[0][0])[i] = 0;
  __syncthreads();

  const int stride = gridDim.x * blockDim.x;
  const int gid = blockIdx.x * blockDim.x + threadIdx.x;
  const int n4 = n >> 2;
  const float4* __restrict__ a4 = (const float4*)arr;

  for (int i = gid; i < n4; i += stride) {
    float4 v = a4[i];                       // global_load_b128
    atomicAdd(&h[wave][clamp_bin(v.x)], 1); // ds_add_u32 (per-wave private)
    atomicAdd(&h[wave][clamp_bin(v.y)], 1);
    atomicAdd(&h[wave][clamp_bin(v.z)], 1);
    atomicAdd(&h[wave][clamp_bin(v.w)], 1);
  }
  for (int i = (n4 << 2) + gid; i < n; i += stride) {
    atomicAdd(&h[wave][clamp_bin(arr[i])], 1);
  }
  __syncthreads();

  for (int i = threadIdx.x; i < NBINS; i += blockDim.x) {
    int s = 0;
#pragma unroll
    for (int w = 0; w < WAVES_PER_BLOCK; ++w) s += h[w][i];
    if (s) atomicAdd(&bins[i], s);          // global_atomic_add_u32
  }
}

// Wave32 full-sum via V_WMMA_F32_16X16X4_F32.
// A VGPR0 = v[lane] -> A[m][0]=v[m] (lanes 0-15), A[m][2]=v[m+16] (lanes 16-31);
// A VGPR1 = 0 -> K=1,3 contribute nothing. B = all ones (any layout works).
// D[m][n] = v[m] + v[m+16] = s[m]. D VGPR j holds s[j] (lanes 0-15) and
// s[j+8] (lanes 16-31); per-lane sum of the 8 D regs + shfl_xor(16) = total.
__device__ __forceinline__ float wave_sum32(float v) {
  v2f a; a.x = v;    a.y = 0.0f;
  v2f b; b.x = 1.0f; b.y = 1.0f;
  v8f c = {};
  c = __builtin_amdgcn_wmma_f32_16x16x4_f32(
      /*neg_a=*/false, a, /*neg_b=*/false, b,
      /*c_mod=*/(short)0, c, /*reuse_a=*/false, /*reuse_b=*/false);
  float q = c[0] + c[1] + c[2] + c[3] + c[4] + c[5] + c[6] + c[7];
  q += __shfl_xor(q, 16, 32);
  return q;
}

__global__ void sim_kernel(const int* __restrict__ bins,
                           const float* __restrict__ depart,
                           const float* __restrict__ initial,
                           const float* __restrict__ finale,
                           float* __restrict__ out) {
  const int lane = threadIdx.x;          // 0..31, one full wave, EXEC all-ones
  const float dep  = depart[0];
  const float ini  = initial[0];
  const float fin  = finale[0];
  const float finm = fin - 0.001f;
  const float P = 8.0f, U = 0.6f, DECAY = 0.06f;

  float cnt[4], soc[4], binf[4];
#pragma unroll
  for (int j = 0; j < 4; ++j) {
    int b = lane + 32 * j;
    cnt[j]  = (float)bins[b];
    soc[j]  = ini;
    binf[j] = (float)b;
  }

  float u_mat = 0.0f;
  for (int t = 0; t < T_STEPS - 1; ++t) {
    const float tf = (float)t;
    // n_active: present & (soc <= final - 0.001), cnt-weighted; exact in f32.
    float ns = 0.0f;
#pragma unroll
    for (int j = 0; j < 4; ++j) {
      bool sel     = binf[j] < tf;       // sorted arrivals: prefix == arrival<t
      bool present = sel && (dep >= tf);
      ns += (present && (soc[j] <= finm)) ? cnt[j] : 0.0f;
    }
    float n_tot = wave_sum32(ns);
    n_tot = fmaxf(n_tot, 1.0f);
    const float shared = P / n_tot;

    float su = 0.0f;
#pragma unroll
    for (int j = 0; j < 4; ++j) {
      bool sel     = binf[j] < tf;
      bool present = sel && (dep >= tf);
      float uval = (present && (soc[j] <= fin)) ? shared : 0.0f;
      float upd  = fminf(fminf(uval, U - DECAY * soc[j]), fin - soc[j]);
      upd = sel ? upd : 0.0f;            // reference masks by sel (t>=t0 implied)
      soc[j] += upd;
      su += cnt[j] * upd;
    }
    u_mat += wave_sum32(su);
  }

  if (lane == 0) out[0] = -u_mat;
}

extern "C" void kernel_launch(void* const* d_in, const int* in_sizes, int n_in,
                              void* d_out, int out_size, void* d_ws, size_t ws_size,
                              hipStream_t stream) {
  const float* arrival = (const float*)d_in[0];
  const float* depart  = (const float*)d_in[1];
  const float* initial = (const float*)d_in[2];
  const float* finale  = (const float*)d_in[3];
  const int n = in_sizes[0];
  int* bins = (int*)d_ws;

  zero_bins_kernel<<<1, NBINS, 0, stream>>>(bins);
  hist_kernel<<<HIST_GRID, HIST_BLOCK, 0, stream>>>(arrival, n, bins);
  sim_kernel<<<1, 32, 0, stream>>>(bins, depart, initial, finale, (float*)d_out);
}